// QML_Model_48859547959352
// MI455X (gfx1250) — compile-verified
//
#include <hip/hip_runtime.h>
#include <math.h>

typedef float v2f __attribute__((ext_vector_type(2)));
typedef float v8f __attribute__((ext_vector_type(8)));

#define EPSC 1e-5f

// ---------------- Phase A: the tiny network, one workgroup ----------------
constexpr int E_ = 22, A_ = 11, P_ = E_ * A_, NL = 6;

// shared-memory layout (floats)
constexpr int OFF_SEQ   = 0;                 // 1936  seq[22][11][8]
constexpr int OFF_RAE   = 1936;              // 242
constexpr int OFF_AEINV = 2178;              // 8
constexpr int OFF_SCAL  = 2186;              // 8
constexpr int OFF_Q     = 2194;              // 1936 (attention scratch)
constexpr int OFF_K     = 4130;              // 1936
constexpr int OFF_V     = 6066;              // 1936
constexpr int OFF_SC    = 8002;              // 2662 -> ends 10664
constexpr int OFF_XA    = 2194;              // 4224 (reuses Q/K/V region)
constexpr int OFF_XB    = 6418;              // 4224
constexpr int OFF_Y     = 10664;             // 352
constexpr int OFF_AR    = 11016;             // 22
constexpr int OFF_X2A   = 11040;             // 384
constexpr int OFF_X2B   = 11424;             // 384
constexpr int OFF_Y2    = 11808;             // 16
constexpr int SMEM_FLOATS = 11824;           // ~47.3 KB

__global__ __launch_bounds__(256) void qml_phaseA(
    const float* __restrict__ pos_a, const float* __restrict__ rpos_w,
    const float* __restrict__ embed_w, const float* __restrict__ embed_b,
    const float* __restrict__ norm_g, const float* __restrict__ norm_b,
    const float* __restrict__ amp_out,
    const float* __restrict__ wq, const float* __restrict__ bq,
    const float* __restrict__ wk, const float* __restrict__ bk,
    const float* __restrict__ wv, const float* __restrict__ bv,
    const float* __restrict__ lin_int_w, const float* __restrict__ lin_int_b,
    const float* __restrict__ nig, const float* __restrict__ nib,
    const float* __restrict__ conv_a_w, const float* __restrict__ conv_e_w,
    const int* __restrict__ pos_ix, const int* __restrict__ atom_ix,
    const int* __restrict__ ix_a, float* __restrict__ ws)
{
    __shared__ float sm[SMEM_FLOATS];
    float* s_seq   = sm + OFF_SEQ;
    float* s_rae   = sm + OFF_RAE;
    float* s_aeinv = sm + OFF_AEINV;
    float* s_scal  = sm + OFF_SCAL;
    float* s_q     = sm + OFF_Q;
    float* s_k     = sm + OFF_K;
    float* s_v     = sm + OFF_V;
    float* s_sc    = sm + OFF_SC;
    float* s_xa    = sm + OFF_XA;
    float* s_xb    = sm + OFF_XB;
    float* s_y     = sm + OFF_Y;
    float* s_ar    = sm + OFF_AR;
    float* s_x2a   = sm + OFF_X2A;
    float* s_x2b   = sm + OFF_X2B;
    float* s_y2    = sm + OFF_Y2;
    const int t = threadIdx.x;

    // ---- embed: seq[e,a,:] = [ae, r] @ embed_w.T + embed_b
    for (int p = t; p < P_; p += 256) {
        int e = p / A_, a = p % A_;
        float pe0 = rpos_w[pos_ix[e]*3+0] + pos_a[atom_ix[e]*3+0];
        float pe1 = rpos_w[pos_ix[e]*3+1] + pos_a[atom_ix[e]*3+1];
        float pe2 = rpos_w[pos_ix[e]*3+2] + pos_a[atom_ix[e]*3+2];
        float d0 = pe0 - pos_a[a*3+0];
        float d1 = pe1 - pos_a[a*3+1];
        float d2 = pe2 - pos_a[a*3+2];
        float r  = sqrtf(d0*d0 + d1*d1 + d2*d2);
        s_rae[p] = r;
        for (int o = 0; o < 8; ++o)
            s_seq[p*8+o] = d0*embed_w[o*4+0] + d1*embed_w[o*4+1] +
                           d2*embed_w[o*4+2] + r *embed_w[o*4+3] + embed_b[o];
    }
    __syncthreads();

    if (t == 0) {  // amp_ae (ddof=1 std), bias_ae (mean) over all 242
        float s = 0.f;
        for (int p = 0; p < P_; ++p) s += s_rae[p];
        float mu = s / P_, ss = 0.f;
        for (int p = 0; p < P_; ++p) { float d = s_rae[p]-mu; ss += d*d; }
        s_scal[0] = sqrtf(ss / (P_-1));
        s_scal[1] = mu;
    } else if (t == 32) {  // last row of pinv(embed_w): Gauss-Jordan 4x4
        float M[4][8];
        for (int i = 0; i < 4; ++i)
            for (int j = 0; j < 4; ++j) {
                float s = 0.f;
                for (int o = 0; o < 8; ++o) s += embed_w[o*4+i]*embed_w[o*4+j];
                M[i][j] = s; M[i][4+j] = (i==j) ? 1.f : 0.f;
            }
        for (int c = 0; c < 4; ++c) {
            int piv = c; float best = fabsf(M[c][c]);
            for (int r2 = c+1; r2 < 4; ++r2) { float v = fabsf(M[r2][c]); if (v > best){best=v;piv=r2;} }
            if (piv != c) for (int j = 0; j < 8; ++j) { float tmp=M[c][j]; M[c][j]=M[piv][j]; M[piv][j]=tmp; }
            float ip = 1.f / M[c][c];
            for (int j = 0; j < 8; ++j) M[c][j] *= ip;
            for (int r2 = 0; r2 < 4; ++r2) if (r2 != c) {
                float f = M[r2][c];
                for (int j = 0; j < 8; ++j) M[r2][j] -= f*M[c][j];
            }
        }
        for (int o = 0; o < 8; ++o) {
            float s = 0.f;
            for (int j = 0; j < 4; ++j) s += M[3][4+j]*embed_w[o*4+j];
            s_aeinv[o] = s;
        }
    }
    __syncthreads();

    // ---- 6 quantum-fidelity attention layers
    for (int L = 0; L < NL; ++L) {
        const float *Wq = wq + L*64, *Bq = bq + L*8;
        const float *Wk = wk + L*64, *Bk = bk + L*8;
        const float *Wv = wv + L*64, *Bv = bv + L*8;
        for (int p = t; p < P_; p += 256) {
            int a = p % A_;
            float amp = (float)ix_a[a];
            float qi[8];
            for (int i = 0; i < 8; ++i) {
                float th = tanhf(s_seq[p*8+i]);
                th = fminf(fmaxf(th, -1.f + EPSC), 1.f - EPSC);
                qi[i] = amp * acosf(th);
            }
            for (int o = 0; o < 8; ++o) {
                float aq = Bq[o], ak = Bk[o], av = Bv[o];
                for (int i = 0; i < 8; ++i) {
                    aq += qi[i]*Wq[o*8+i]; ak += qi[i]*Wk[o*8+i]; av += qi[i]*Wv[o*8+i];
                }
                s_q[p*8+o] = aq; s_k[p*8+o] = ak; s_v[p*8+o] = av;
            }
        }
        __syncthreads();
        for (int idx = t; idx < E_*A_*A_; idx += 256) {
            int e = idx/(A_*A_), rem = idx%(A_*A_), i = rem/A_, j = rem%A_;
            const float* qp = s_q + (e*A_+i)*8;
            const float* kp = s_k + (e*A_+j)*8;
            float pr = 1.f;
            for (int d = 0; d < 8; ++d) { float c = cosf(0.5f*(qp[d]-kp[d])); pr *= c*c; }
            s_sc[idx] = pr;
        }
        __syncthreads();
        for (int p = t; p < P_; p += 256) {
            int e = p / A_, i = p % A_;
            const float* row = s_sc + (e*A_+i)*A_;
            float ssum = 1e-9f;
            for (int j = 0; j < A_; ++j) ssum += row[j];
            float inv = 1.f / ssum;
            float tmp[8];
            float mu = 0.f;
            for (int o = 0; o < 8; ++o) {
                float acc = 0.f;
                for (int j = 0; j < A_; ++j) acc += row[j]*s_v[(e*A_+j)*8+o];
                tmp[o] = s_seq[p*8+o] + acc*inv;
                mu += tmp[o];
            }
            mu *= 0.125f;
            float var = 0.f;
            for (int o = 0; o < 8; ++o) { float d = tmp[o]-mu; var += d*d; }
            var *= 0.125f;
            float rs = rsqrtf(var + EPSC);
            for (int o = 0; o < 8; ++o)
                s_seq[p*8+o] = (tmp[o]-mu)*rs*norm_g[o] + norm_b[o];
        }
        __syncthreads();
    }

    // ---- r_ae from pseudo-inverse row, re-standardize
    for (int p = t; p < P_; p += 256) {
        float r = 0.f;
        for (int o = 0; o < 8; ++o) r += s_seq[p*8+o]*s_aeinv[o];
        s_rae[p] = r;
    }
    __syncthreads();
    if (t == 0) {
        float s = 0.f;
        for (int p = 0; p < P_; ++p) s += s_rae[p];
        float mu = s / P_, ss = 0.f;
        for (int p = 0; p < P_; ++p) { float d = s_rae[p]-mu; ss += d*d; }
        s_scal[2] = mu;
        s_scal[3] = sqrtf(ss / (P_-1));
    }
    __syncthreads();
    {
        float amp_ae = s_scal[0], bias_ae = s_scal[1], mu2 = s_scal[2], sd2 = s_scal[3];
        for (int p = t; p < P_; p += 256)
            s_rae[p] = amp_ae*(s_rae[p]-mu2)/sd2 + bias_ae;
    }
    __syncthreads();

    // ---- lin_int, transposed into conv layout [e][16][12]
    float amp_out0 = amp_out[0];
    for (int p = t; p < P_; p += 256) {
        int e = p / A_, a = p % A_;
        float sc = expf(-s_rae[p] + amp_out0) * (float)ix_a[a];
        float v8[8];
        for (int i = 0; i < 8; ++i) v8[i] = sc * s_seq[p*8+i];
        for (int o = 0; o < 16; ++o) {
            float acc = lin_int_b[o];
            for (int i = 0; i < 8; ++i) acc += v8[i]*lin_int_w[o*8+i];
            s_xa[(e*16+o)*12 + a] = acc;
        }
    }
    __syncthreads();
    for (int idx = t; idx < E_*16; idx += 256) {
        int e = idx/16, o = idx%16;
        float acc = 0.f;
        for (int a = 0; a < A_; ++a) acc += s_xa[(e*16+o)*12+a];
        s_y[idx] = acc / 11.f;
    }
    for (int e = t; e < E_; e += 256) {
        float acc = 0.f;
        for (int a = 0; a < A_; ++a) acc += expf(-s_rae[e*A_+a]);
        s_ar[e] = acc / 11.f;
    }
    __syncthreads();

    // ---- conv_a pyramid: 11 -> 6 -> 3 -> 2 -> 1 -> 1 -> 1  (6 iters)
    float* src = s_xa; float* dst = s_xb; int Lc = 11;
    for (int it = 0; it < 6; ++it) {
        int Lo = (Lc+1) >> 1;
        int tot = E_*16*Lo;
        for (int idx = t; idx < tot; idx += 256) {
            int e = idx/(16*Lo), rem = idx%(16*Lo), o = rem/Lo, lp = rem%Lo;
            float acc = 0.f;
            int p0 = 2*lp, p1 = 2*lp+1;
            for (int i = 0; i < 16; ++i) {
                int base = (e*16+i)*12;
                float x0 = (p0 < Lc) ? src[base+p0] : 0.f;
                float x1 = (p1 < Lc) ? src[base+p1] : 0.f;
                acc += conv_a_w[(o*16+i)*2+0]*x0 + conv_a_w[(o*16+i)*2+1]*x1;
            }
            dst[(e*16+o)*12+lp] = acc;
        }
        __syncthreads();
        float* tp = src; src = dst; dst = tp; Lc = Lo;
    }

    // ---- per-e layernorm(16), *amp_r, transpose into [16][24]
    for (int e = t; e < E_; e += 256) {
        float z[16]; float mu = 0.f;
        for (int o = 0; o < 16; ++o) { z[o] = s_y[e*16+o] + src[(e*16+o)*12]; mu += z[o]; }
        mu *= (1.f/16.f);
        float var = 0.f;
        for (int o = 0; o < 16; ++o) { float d = z[o]-mu; var += d*d; }
        var *= (1.f/16.f);
        float rs = rsqrtf(var + EPSC);
        float ar = s_ar[e];
        for (int o = 0; o < 16; ++o)
            s_x2a[o*24+e] = ar*((z[o]-mu)*rs*nig[o] + nib[o]);
    }
    __syncthreads();
    for (int o = t; o < 16; o += 256) {
        float acc = 0.f;
        for (int e = 0; e < E_; ++e) acc += s_x2a[o*24+e];
        s_y2[o] = acc / 22.f;
    }
    if (t == 0) {
        float acc = 0.f;
        for (int e = 0; e < E_; ++e) acc += s_ar[e];
        s_scal[4] = acc / 22.f;   // amp_r2
    }
    __syncthreads();

    // ---- conv_e pyramid: 22 -> 11 -> 6 -> 3 -> 2 -> 1 ... (11 iters)
    src = s_x2a; dst = s_x2b; Lc = 22;
    for (int it = 0; it < 11; ++it) {
        int Lo = (Lc+1) >> 1;
        int tot = 16*Lo;
        for (int idx = t; idx < tot; idx += 256) {
            int o = idx/Lo, lp = idx%Lo;
            int p0 = 2*lp, p1 = 2*lp+1;
            float acc = 0.f;
            for (int i = 0; i < 16; ++i) {
                float x0 = (p0 < Lc) ? src[i*24+p0] : 0.f;
                float x1 = (p1 < Lc) ? src[i*24+p1] : 0.f;
                acc += conv_e_w[(o*16+i)*2+0]*x0 + conv_e_w[(o*16+i)*2+1]*x1;
            }
            dst[o*24+lp] = acc;
        }
        __syncthreads();
        float* tp = src; src = dst; dst = tp; Lc = Lo;
    }

    if (t == 0) {
        float z[16]; float mu = 0.f;
        for (int o = 0; o < 16; ++o) { z[o] = s_y2[o] + src[o*24]; mu += z[o]; }
        mu *= (1.f/16.f);
        float var = 0.f;
        for (int o = 0; o < 16; ++o) { float d = z[o]-mu; var += d*d; }
        var *= (1.f/16.f);
        float rs = rsqrtf(var + EPSC);
        float ar2 = s_scal[4];
        for (int o = 0; o < 16; ++o)
            ws[o] = ar2*((z[o]-mu)*rs*nig[o] + nib[o]);   // final x16 -> workspace
    }
}

// ---------------- Phase B: Kronecker half-tables for bos ----------------
// bos(i) = hi[i>>11] * lo[i&2047];  hi covers qubits 0..10, lo covers 11..21
__global__ void qml_bos_tables(const float* __restrict__ hf, float* __restrict__ ws)
{
    int i = blockIdx.x*blockDim.x + threadIdx.x;
    if (i >= 4096) return;
    int half = i >> 11;
    int idx  = i & 2047;
    int qb   = half * 11;
    float pr = 1.f;
    for (int j = 0; j < 11; ++j) {
        float th = 0.5f * hf[qb + j];
        int bit = (idx >> (10-j)) & 1;
        pr *= bit ? sinf(th) : cosf(th);
    }
    ws[16 + half*2048 + idx] = pr;
}

// ---------------- Phase C: memory-bound GEMV via V_WMMA_F32_16X16X4_F32 ----------------
// Per 16-row tile: A(16x4) = x-chunk broadcast (lanes 0-15: K0/K1, lanes 16-31: K2/K3),
// B(4x16) = W tile (lane n%16 holds col n, VGPR0/1 = K pair) -> D[m,n] = out[row n] in all lanes.
__global__ __launch_bounds__(256) void qml_gemv_wmma(
    const float* __restrict__ W, const float* __restrict__ bias,
    const float* __restrict__ ws, float* __restrict__ out, int ntiles)
{
    const int lane = threadIdx.x & 31;
    const int wid  = blockIdx.x * (blockDim.x >> 5) + (threadIdx.x >> 5);
    const int nw   = gridDim.x * (blockDim.x >> 5);
    const int r    = lane & 15;
    const bool hiH = lane >= 16;

    v2f a0, a1, a2, a3;
    {
        const int off = hiH ? 2 : 0;
        a0.x = ws[ 0+off]; a0.y = ws[ 1+off];
        a1.x = ws[ 4+off]; a1.y = ws[ 5+off];
        a2.x = ws[ 8+off]; a2.y = ws[ 9+off];
        a3.x = ws[12+off]; a3.y = ws[13+off];
    }
    const float* hi = ws + 16;
    const float* lo = ws + 16 + 2048;

    for (int tile = wid; tile < ntiles; tile += nw) {
        const int row0 = tile << 4;
        const float* wrow = W + (size_t)(row0 + r)*16 + (hiH ? 2 : 0);
        v2f b0 = *(const v2f*)(wrow);
        v2f b1 = *(const v2f*)(wrow + 4);
        v2f b2 = *(const v2f*)(wrow + 8);
        v2f b3 = *(const v2f*)(wrow + 12);
        v8f d = {0.f,0.f,0.f,0.f,0.f,0.f,0.f,0.f};
        d = __builtin_amdgcn_wmma_f32_16x16x4_f32(false, a0, false, b0, (short)0, d, false, false);
        d = __builtin_amdgcn_wmma_f32_16x16x4_f32(false, a1, false, b1, (short)0, d, false, false);
        d = __builtin_amdgcn_wmma_f32_16x16x4_f32(false, a2, false, b2, (short)0, d, false, false);
        d = __builtin_amdgcn_wmma_f32_16x16x4_f32(false, a3, false, b3, (short)0, d, false, false);
        if (!hiH) {
            const int row = row0 + r;
            float bv = hi[row >> 11] * lo[row & 2047];
            out[row] = d[0] + bias[row] + bv * 2048.0f;   // 2^(22/2)
        }
    }
}

extern "C" void kernel_launch(void* const* d_in, const int* in_sizes, int n_in,
                              void* d_out, int out_size, void* d_ws, size_t ws_size,
                              hipStream_t stream) {
    const float* pos_a     = (const float*)d_in[0];
    const float* hf        = (const float*)d_in[1];
    const float* rpos_w    = (const float*)d_in[2];
    const float* embed_w   = (const float*)d_in[3];
    const float* embed_b   = (const float*)d_in[4];
    const float* norm_g    = (const float*)d_in[5];
    const float* norm_b    = (const float*)d_in[6];
    const float* amp_out   = (const float*)d_in[7];
    const float* qsa_wq    = (const float*)d_in[8];
    const float* qsa_bq    = (const float*)d_in[9];
    const float* qsa_wk    = (const float*)d_in[10];
    const float* qsa_bk    = (const float*)d_in[11];
    const float* qsa_wv    = (const float*)d_in[12];
    const float* qsa_bv    = (const float*)d_in[13];
    const float* lin_int_w = (const float*)d_in[14];
    const float* lin_int_b = (const float*)d_in[15];
    const float* nig       = (const float*)d_in[16];
    const float* nib       = (const float*)d_in[17];
    const float* conv_a_w  = (const float*)d_in[18];
    const float* conv_e_w  = (const float*)d_in[19];
    const float* lin_out_w = (const float*)d_in[20];
    const float* lin_out_b = (const float*)d_in[21];
    const int*   pos_ix    = (const int*)d_in[22];
    const int*   atom_ix   = (const int*)d_in[23];
    const int*   ix_a      = (const int*)d_in[24];
    float* out = (float*)d_out;
    float* ws  = (float*)d_ws;

    qml_phaseA<<<1, 256, 0, stream>>>(
        pos_a, rpos_w, embed_w, embed_b, norm_g, norm_b, amp_out,
        qsa_wq, qsa_bq, qsa_wk, qsa_bk, qsa_wv, qsa_bv,
        lin_int_w, lin_int_b, nig, nib, conv_a_w, conv_e_w,
        pos_ix, atom_ix, ix_a, ws);

    qml_bos_tables<<<16, 256, 0, stream>>>(hf, ws);

    const int ntiles = (out_size + 15) / 16;   // 2^22 / 16 = 262144
    qml_gemv_wmma<<<2048, 256, 0, stream>>>(lin_out_w, lin_out_b, ws, out, ntiles);
}